// QB_71502615544514
// MI455X (gfx1250) — compile-verified
//
#include <hip/hip_runtime.h>
#include <math.h>

typedef __attribute__((ext_vector_type(2))) float v2f;
typedef __attribute__((ext_vector_type(4))) float v4f;
typedef __attribute__((ext_vector_type(8))) float v8f;

#define EPS 1e-5f

// Single-instruction static lane shuffle: ds_swizzle_b32 (group-of-32 mode).
// IMM: and_mask=[4:0], or_mask=[9:5], xor_mask=[14:10]; src lane = ((i&and)|or)^xor
template <int IMM>
__device__ __forceinline__ float ds_swz(float v) {
  return __int_as_float(__builtin_amdgcn_ds_swizzle(__float_as_int(v), IMM));
}

// Apply shared 2x2 complex gate (g = re00,im00,re01,im01,re10,im10,re11,im11)
// to the wire whose amplitude-index bit mask is M. Fully unrolled: constant indices.
template <int M>
__device__ __forceinline__ void apply_gate(float re[16], float im[16], const float g[8]) {
#pragma unroll
  for (int i0 = 0; i0 < 16; ++i0) {
    if (i0 & M) continue;
    const int i1 = i0 | M;
    const float a0r = re[i0], a0i = im[i0];
    const float a1r = re[i1], a1i = im[i1];
    re[i0] = g[0] * a0r - g[1] * a0i + g[2] * a1r - g[3] * a1i;
    im[i0] = g[0] * a0i + g[1] * a0r + g[2] * a1i + g[3] * a1r;
    re[i1] = g[4] * a0r - g[5] * a0i + g[6] * a1r - g[7] * a1i;
    im[i1] = g[4] * a0i + g[5] * a0r + g[6] * a1i + g[7] * a1r;
  }
}

// CNOT: control mask CM, target mask TM. Pure register renames after unrolling.
template <int CM, int TM>
__device__ __forceinline__ void cnot16(float re[16], float im[16]) {
#pragma unroll
  for (int i = 0; i < 16; ++i) {
    if ((i & CM) && !(i & TM)) {
      const int j = i | TM;
      float t = re[i]; re[i] = re[j]; re[j] = t;
      t = im[i]; im[i] = im[j]; im[j] = t;
    }
  }
}

// One 16-sample M-tile: gather A frag (ds_swizzle ASWZ), 4x WMMA over the 64
// output channels, fused LayerNorm via ds_swizzle XOR butterflies, NT stores.
template <int ASWZ>
__device__ __forceinline__ void tile_gemm_ln(
    const float q[4], const v2f Bf[4], const float bsv[4],
    const float gam[4], const float bet[4],
    int half, int nlo, int tileRow0,
    float* __restrict__ out, int B) {
  // A layout (16x4 f32): lanes 0-15 hold K=0,1; lanes 16-31 hold K=2,3; M=lane&15
  const float a0lo = ds_swz<ASWZ>(q[0]);
  const float a1lo = ds_swz<ASWZ>(q[1]);
  const float a0hi = ds_swz<ASWZ>(q[2]);
  const float a1hi = ds_swz<ASWZ>(q[3]);
  v2f A;
  A.x = half ? a0hi : a0lo;
  A.y = half ? a1hi : a1lo;

  v8f acc[4];
#pragma unroll
  for (int gi = 0; gi < 4; ++gi) {
    v8f c = {};  // inline-0 C operand
    acc[gi] = __builtin_amdgcn_wmma_f32_16x16x4_f32(
        false, A, false, Bf[gi], (short)0, c, false, false);
#pragma unroll
    for (int r = 0; r < 8; ++r) acc[gi][r] += bsv[gi];  // h = qW^T + b
  }

  // LayerNorm over 64 channels, directly on the WMMA D layout:
  // row M = r + 8*half lives in VGPR r across one 16-lane half (all 4 groups).
#pragma unroll
  for (int r = 0; r < 8; ++r) {
    float s1 = acc[0][r] + acc[1][r] + acc[2][r] + acc[3][r];
    float s2 = acc[0][r] * acc[0][r] + acc[1][r] * acc[1][r] +
               acc[2][r] * acc[2][r] + acc[3][r] * acc[3][r];
    // XOR butterflies (masks 1,2,4,8) stay within each 16-lane half.
    s1 += ds_swz<0x041f>(s1);  s2 += ds_swz<0x041f>(s2);
    s1 += ds_swz<0x081f>(s1);  s2 += ds_swz<0x081f>(s2);
    s1 += ds_swz<0x101f>(s1);  s2 += ds_swz<0x101f>(s2);
    s1 += ds_swz<0x201f>(s1);  s2 += ds_swz<0x201f>(s2);

    const float mu  = s1 * 0.015625f;                 // /64
    const float var = s2 * 0.015625f - mu * mu;
    const float rs  = __frsqrt_rn(var + EPS);

    const int row = tileRow0 + r + 8 * half;
    if (row < B) {
      float* orow = out + (size_t)row * 64;
#pragma unroll
      for (int gi = 0; gi < 4; ++gi) {
        const float v = gam[gi] * (acc[gi][r] - mu) * rs + bet[gi];
        __builtin_nontemporal_store(v, orow + 16 * gi + nlo);
      }
    }
  }
}

__global__ __launch_bounds__(256) void QB_qnn_wmma_kernel(
    const float* __restrict__ x,        // [B,4]
    const float* __restrict__ weights,  // [2,4,3]
    const float* __restrict__ W,        // [64,4]
    const float* __restrict__ bias,     // [64]
    const float* __restrict__ gamma,    // [64]
    const float* __restrict__ beta,     // [64]
    float* __restrict__ out,            // [B,64]
    int B) {
  // ---- shared Rot gate matrices (batch-invariant), computed once per block ----
  __shared__ float gmat[8][8];
  const int tid = threadIdx.x;
  if (tid < 8) {
    const float phi = weights[tid * 3 + 0];
    const float th  = weights[tid * 3 + 1];
    const float om  = weights[tid * 3 + 2];
    float s, c, spo, cpo, smo, cmo;
    __sincosf(0.5f * th, &s, &c);
    __sincosf(0.5f * (phi + om), &spo, &cpo);
    __sincosf(0.5f * (phi - om), &smo, &cmo);
    gmat[tid][0] =  cpo * c;  gmat[tid][1] = -spo * c;   // m00 = e^{-i(phi+om)/2} c
    gmat[tid][2] = -cmo * s;  gmat[tid][3] = -smo * s;   // m01 = -e^{+i(phi-om)/2} s
    gmat[tid][4] =  cmo * s;  gmat[tid][5] = -smo * s;   // m10 = e^{-i(phi-om)/2} s
    gmat[tid][6] =  cpo * c;  gmat[tid][7] =  spo * c;   // m11 = e^{+i(phi+om)/2} c
  }
  __syncthreads();

  // ---- per-lane quantum circuit (one sample per lane) ----
  const int sample = blockIdx.x * 256 + tid;
  const int sload  = sample < B ? sample : B - 1;
  const v4f xv = __builtin_nontemporal_load(
      reinterpret_cast<const v4f*>(x) + sload);

  const float PIH = 1.57079632679489662f;  // pi/2 (RX half-angle scale)
  float cw[4], sw[4];
  {
    const float xa[4] = {xv.x, xv.y, xv.z, xv.w};
#pragma unroll
    for (int w = 0; w < 4; ++w) {
      const float ha = tanhf(xa[w]) * PIH;     // angle/2
      __sincosf(ha, &sw[w], &cw[w]);
    }
  }

  // RX embedding as a direct product state: amp = (prod mags) * (-i)^popcount
  float re[16], im[16];
#pragma unroll
  for (int idx = 0; idx < 16; ++idx) {
    float mag = 1.0f;
    int pop = 0;
#pragma unroll
    for (int w = 0; w < 4; ++w) {
      if ((idx >> (3 - w)) & 1) { mag *= sw[w]; ++pop; }
      else                      { mag *= cw[w]; }
    }
    const int ph = pop & 3;
    re[idx] = (ph == 0) ? mag : ((ph == 2) ? -mag : 0.0f);
    im[idx] = (ph == 1) ? -mag : ((ph == 3) ? mag : 0.0f);
  }

  // StronglyEntanglingLayers: 2 layers of Rot + ring CNOTs (r=1 then r=2)
  float g[8];
#define LOADG(row)                         \
  {                                        \
    _Pragma("unroll")                      \
    for (int e = 0; e < 8; ++e) g[e] = gmat[row][e]; \
  }
  LOADG(0) apply_gate<8>(re, im, g);
  LOADG(1) apply_gate<4>(re, im, g);
  LOADG(2) apply_gate<2>(re, im, g);
  LOADG(3) apply_gate<1>(re, im, g);
  cnot16<8, 4>(re, im); cnot16<4, 2>(re, im);
  cnot16<2, 1>(re, im); cnot16<1, 8>(re, im);
  LOADG(4) apply_gate<8>(re, im, g);
  LOADG(5) apply_gate<4>(re, im, g);
  LOADG(6) apply_gate<2>(re, im, g);
  LOADG(7) apply_gate<1>(re, im, g);
  cnot16<8, 2>(re, im); cnot16<4, 1>(re, im);
  cnot16<2, 8>(re, im); cnot16<1, 4>(re, im);
#undef LOADG

  // PauliZ expvals + softmax -> q[4]
  float q[4];
  {
    float p[16];
#pragma unroll
    for (int i = 0; i < 16; ++i) p[i] = re[i] * re[i] + im[i] * im[i];
    float e[4];
#pragma unroll
    for (int w = 0; w < 4; ++w) {
      const int m = 8 >> w;
      float acc = 0.0f;
#pragma unroll
      for (int i = 0; i < 16; ++i) acc += (i & m) ? -p[i] : p[i];
      e[w] = acc;
    }
    float mx = fmaxf(fmaxf(e[0], e[1]), fmaxf(e[2], e[3]));
    float sum = 0.0f;
#pragma unroll
    for (int w = 0; w < 4; ++w) { q[w] = __expf(e[w] - mx); sum += q[w]; }
    const float inv = __frcp_rn(sum);
#pragma unroll
    for (int w = 0; w < 4; ++w) q[w] *= inv;
  }

  // ---- h = q @ W^T + b via V_WMMA_F32_16X16X4_F32, then fused LayerNorm ----
  const int lane = tid & 31;
  const int half = lane >> 4;   // 0: lanes 0-15, 1: lanes 16-31
  const int nlo  = lane & 15;   // N (channel within group) / row within tile

  // B fragment (4x16 slice of W^T per 16-channel group), bias/gamma/beta per lane
  v2f Bf[4];
  float bsv[4], gam[4], bet[4];
#pragma unroll
  for (int gi = 0; gi < 4; ++gi) {
    const v4f wr = reinterpret_cast<const v4f*>(W)[16 * gi + nlo];
    Bf[gi].x = half ? wr.z : wr.x;  // VGPR0: K=0 (low half) / K=2 (high half)
    Bf[gi].y = half ? wr.w : wr.y;  // VGPR1: K=1 / K=3
    bsv[gi] = bias[16 * gi + nlo];
    gam[gi] = gamma[16 * gi + nlo];
    bet[gi] = beta[16 * gi + nlo];
  }

  const int waveBase = blockIdx.x * 256 + (tid & ~31);  // wave's first sample

  // Two 16-sample M-tiles per wave (32 lanes = 32 samples).
  // A gather: src = 16*t + (lane&15) -> and=0x0f (t=0), and|or16 (t=1).
  tile_gemm_ln<0x00f>(q, Bf, bsv, gam, bet, half, nlo, waveBase,      out, B);
  tile_gemm_ln<0x20f>(q, Bf, bsv, gam, bet, half, nlo, waveBase + 16, out, B);
}

extern "C" void kernel_launch(void* const* d_in, const int* in_sizes, int n_in,
                              void* d_out, int out_size, void* d_ws, size_t ws_size,
                              hipStream_t stream) {
  const float* x       = (const float*)d_in[0];
  const float* weights = (const float*)d_in[1];
  const float* W       = (const float*)d_in[2];
  const float* b       = (const float*)d_in[3];
  const float* gamma   = (const float*)d_in[4];
  const float* beta    = (const float*)d_in[5];
  float* out = (float*)d_out;

  const int B = in_sizes[0] / 4;
  const int blocks = (B + 255) / 256;
  hipLaunchKernelGGL(QB_qnn_wmma_kernel, dim3(blocks), dim3(256), 0, stream,
                     x, weights, W, b, gamma, beta, out, B);
}